// FeatTransKhop_14405320311052
// MI455X (gfx1250) — compile-verified
//
#include <hip/hip_runtime.h>

typedef __attribute__((ext_vector_type(2))) float v2f;
typedef __attribute__((ext_vector_type(8))) float v8f;

#define NSAMP   2048
#define KHOP    3
#define NNB     16
#define F       128
#define OUTC    256      // C_OUT * F_OUT
#define ZROWS   64       // 1 (x) + 48 (neighbors) padded to 64
#define ZPITCH  132      // 132 % 64 = 4 -> conflict-free fragment gathers
#define APITCH  132
#define EPS_ROOT 1e-8f
#define EPS_NORM 1e-7f

// static LDS partition (floats)
#define LDS_Z    (ZROWS * ZPITCH)          // 8448
#define LDS_Y    (ZROWS * ZPITCH)          // 8448
#define LDS_ADJ  (F * APITCH)              // 16896
#define LDS_SS   (KHOP * F)                // 384
#define LDS_PS   (256)                     // 256
// total 34432 floats = 137,728 B < 160 KB per CU (CUMODE), < 320 KB per WGP

#define WT_ELEMS (KHOP * 32 * 2 * OUTC * 2)   // 98304 floats in d_ws

// ---------------------------------------------------------------------------
// Pre-transpose W (256 x 3 x 128) into a WMMA-B-fragment-native layout:
//   Wt[K][s][h][o][c] = W[o, K, a]  with  a = 4*s + 2*h + c
// so a GEMM2 B-fragment is one coalesced b64 load per lane.
// ---------------------------------------------------------------------------
__global__ void __launch_bounds__(256)
wtx_transpose_kernel(const float* __restrict__ W, float* __restrict__ Wt) {
    int idx = blockIdx.x * 256 + threadIdx.x;
    if (idx >= WT_ELEMS) return;
    int c = idx & 1;
    int o = (idx >> 1) & 255;
    int h = (idx >> 9) & 1;
    int s = (idx >> 10) & 31;
    int K = idx >> 15;
    int a = 4 * s + 2 * h + c;
    Wt[idx] = W[(o * KHOP + K) * F + a];
}

// ---------------------------------------------------------------------------
// One workgroup (8 wave32) per sample i.
// ---------------------------------------------------------------------------
__global__ void __launch_bounds__(256)
feattrans_kernel(const float* __restrict__ x,
                 const float* __restrict__ nb,
                 const float* __restrict__ Wt,
                 float* __restrict__ out) {
    __shared__ float Zp[LDS_Z];        // [64][132], row 0 == x_i
    __shared__ float Yp[LDS_Y];        // [64][132]
    __shared__ float adjp[LDS_ADJ];    // [128][132], current K
    __shared__ float ss[LDS_SS];       // [3][128] neighbor sums
    __shared__ float psum[LDS_PS];     // [256] partial column abs-sums

    const int i    = blockIdx.x;
    const int tid  = threadIdx.x;
    const int lane = tid & 31;
    const int w    = tid >> 5;         // wave id 0..7
    const int m    = lane & 15;        // fragment row/col index
    const int h    = lane >> 4;        // lane half

    // ---- Phase 0: stage Z = [x ; neighbors ; zeros] into LDS ----
    for (int idx = tid; idx < ZROWS * F; idx += 256) {
        int r = idx >> 7, col = idx & 127;
        float v;
        if (r == 0)       v = x[(size_t)i * F + col];
        else if (r <= 48) {
            int r1 = r - 1;
            v = nb[((size_t)(i * KHOP + (r1 >> 4)) * NNB + (r1 & 15)) * F + col];
        } else            v = 0.f;
        Zp[r * ZPITCH + col] = v;
    }
    __syncthreads();

    // ---- Phase 1: s[k][b] = sum_n neighbor[i,k,n,b] ----
    for (int j = tid; j < KHOP * F; j += 256) {
        int k = j >> 7, b = j & 127;
        float acc = 0.f;
        int base = (1 + k * NNB) * ZPITCH + b;
#pragma unroll
        for (int n = 0; n < NNB; ++n) acc += Zp[base + n * ZPITCH];
        ss[j] = acc;
    }
    __syncthreads();

    // Output accumulators persist across K (64 VGPRs).
    v8f acc2[2][4];
#pragma unroll
    for (int j = 0; j < 2; ++j)
#pragma unroll
        for (int rt = 0; rt < 4; ++rt) acc2[j][rt] = {};

    for (int K = 0; K < KHOP; ++K) {
        // ---- adj_K = row_normalize(sgnroot(x s^T + s x^T)) ----
        {
            int b    = tid & 127;
            int half = tid >> 7;
            float sb = ss[K * F + b];
            float xb = Zp[b];                       // row 0 of Z == x
            float ps = 0.f;
            for (int a = half * 64; a < half * 64 + 64; ++a) {
                float v = Zp[a] * sb + ss[K * F + a] * xb;   // x_a*s_b + s_a*x_b
                float r = sqrtf(fmaxf(fabsf(v), EPS_ROOT));
                float t = (v > 0.f) ? r : ((v < 0.f) ? -r : 0.f);
                adjp[a * APITCH + b] = t;
                ps += fabsf(t);
            }
            psum[tid] = ps;
        }
        __syncthreads();
        for (int idx = tid; idx < F * F; idx += 256) {
            int a = idx >> 7, b = idx & 127;
            float cs = psum[b] + psum[128 + b] + EPS_NORM;
            float v  = adjp[a * APITCH + b] / cs;
            adjp[a * APITCH + b] = (v == v) ? v : 0.f;       // NaN -> 0
        }
        __syncthreads();

        // ---- GEMM1: Y(64x128) = Z(64x128) * adj_K^T ----
        // wave w owns output columns a in [16w, 16w+16)
        {
            const float* Brow = &adjp[(16 * w + m) * APITCH];   // B[k,n]=adj[a,b]
            for (int rt = 0; rt < 4; ++rt) {
                const float* Arow = &Zp[(16 * rt + m) * ZPITCH];
                // two independent accumulator chains to pipeline the matrix unit
                v8f accE = {};
                v8f accO = {};
#pragma unroll
                for (int s = 0; s < 16; ++s) {
                    v2f A0 = *(const v2f*)(Arow + 8 * s + 2 * h);
                    v2f B0 = *(const v2f*)(Brow + 8 * s + 2 * h);
                    v2f A1 = *(const v2f*)(Arow + 8 * s + 4 + 2 * h);
                    v2f B1 = *(const v2f*)(Brow + 8 * s + 4 + 2 * h);
                    accE = __builtin_amdgcn_wmma_f32_16x16x4_f32(
                        false, A0, false, B0, (short)0, accE, false, false);
                    accO = __builtin_amdgcn_wmma_f32_16x16x4_f32(
                        false, A1, false, B1, (short)0, accO, false, false);
                }
                v8f acc = accE + accO;
#pragma unroll
                for (int v = 0; v < 8; ++v)
                    Yp[(16 * rt + v + 8 * h) * ZPITCH + 16 * w + m] = acc[v];
            }
        }
        __syncthreads();

        // ---- GEMM2: Out(64x256) += Y(64x128) * W_K^T  (B from d_ws, L2-hot) ----
#pragma unroll
        for (int j = 0; j < 2; ++j) {
            int ot = w + 8 * j;                                 // o-tile 0..15
            const float* wbase = Wt + (size_t)((K * 32) * 2 + h) * (OUTC * 2)
                                    + (16 * ot + m) * 2;
            for (int s = 0; s < 32; ++s) {
                v2f B = *(const v2f*)(wbase + (size_t)s * (2 * OUTC * 2));
#pragma unroll
                for (int rt = 0; rt < 4; ++rt) {               // 4 independent chains
                    v2f A = *(const v2f*)(&Yp[(16 * rt + m) * ZPITCH] + 4 * s + 2 * h);
                    acc2[j][rt] = __builtin_amdgcn_wmma_f32_16x16x4_f32(
                        false, A, false, B, (short)0, acc2[j][rt], false, false);
                }
            }
        }
        __syncthreads();   // protect Yp/adjp before next K overwrites them
    }

    // ---- Phase 3: scatter accumulators to x_out / nbr_out ----
    const size_t X0 = (size_t)NSAMP * OUTC;   // nbr_out offset in flat d_out
#pragma unroll
    for (int j = 0; j < 2; ++j) {
        int ot  = w + 8 * j;
        int col = 16 * ot + m;                // o = c*128 + f
#pragma unroll
        for (int rt = 0; rt < 4; ++rt) {
#pragma unroll
            for (int v = 0; v < 8; ++v) {
                int row = 16 * rt + v + 8 * h;
                float val = acc2[j][rt][v];
                if (row == 0) {
                    out[(size_t)i * OUTC + col] = val;
                } else if (row <= 48) {
                    int r1 = row - 1;
                    out[X0 + ((size_t)(i * KHOP + (r1 >> 4)) * NNB + (r1 & 15)) * OUTC + col] = val;
                } // rows 49..63 are padding, dropped
            }
        }
    }
}

// ---------------------------------------------------------------------------
extern "C" void kernel_launch(void* const* d_in, const int* in_sizes, int n_in,
                              void* d_out, int out_size, void* d_ws, size_t ws_size,
                              hipStream_t stream) {
    (void)in_sizes; (void)n_in; (void)out_size; (void)ws_size;
    const float* x  = (const float*)d_in[0];
    const float* nb = (const float*)d_in[1];
    const float* W  = (const float*)d_in[2];
    float* outp = (float*)d_out;
    float* Wt   = (float*)d_ws;   // 98304 floats = 384 KB, rewritten every call

    wtx_transpose_kernel<<<WT_ELEMS / 256, 256, 0, stream>>>(W, Wt);
    feattrans_kernel<<<NSAMP, 256, 0, stream>>>(x, nb, Wt, outp);
}